// OcclusionAwareGrouping_70351564309211
// MI455X (gfx1250) — compile-verified
//
#include <hip/hip_runtime.h>
#include <math.h>
#include <float.h>

// MI455X (gfx1250). HBM-bound (227MB input, ~4 GFLOP): two passes over x at
// 23.3 TB/s ~= 20us floor. fp32 end-to-end, both GEMMs on V_WMMA_F32_16X16X4_F32,
// patch tiles staged with double-buffered GLOBAL_LOAD_ASYNC_TO_LDS_B128
// (ASYNCcnt) so tile i+1's HBM fetch overlaps tile i's WMMA work.

typedef float v2f __attribute__((ext_vector_type(2)));
typedef float v8f __attribute__((ext_vector_type(8)));

#define BB   128
#define NN   577
#define PP   576
#define DD   768
#define GG   17
#define KOCC 10
#define INVTEMP 10.0f
#define AS   788            // LDS row stride (788 % 64 == 20): bank-conflict-free
                            // for GEMM1 A-frags and GEMM2 B-frags
#define TILEF (16 * AS)     // floats per tile buffer
#define FEATS (BB*18*DD)

// ------------------------------------------------------------------
// Kernel 1: cls copy, cosine sim (wave-per-patch), top-10 min -> mask
// ------------------------------------------------------------------
__global__ __launch_bounds__(256)
void prep_kernel(const float* __restrict__ x, float* __restrict__ out,
                 float* __restrict__ maskws) {
  __shared__ float clsL[DD];
  __shared__ float simv[PP];
  __shared__ float maskv[PP];
  __shared__ float redv[256];
  __shared__ int   redi[256];
  const int b = blockIdx.x;
  const int t = threadIdx.x;
  const int lane = t & 31, wv = t >> 5;
  const float* xb = x + (size_t)b * NN * DD;

  for (int d = t; d < DD; d += 256) {
    float v = xb[d];
    clsL[d] = v;
    out[(size_t)b * 18 * DD + d] = v;      // feats row 0 = cls token
  }
  for (int p = t; p < PP; p += 256) maskv[p] = 1.0f;
  __syncthreads();

  // sim[p] = <patch_p, cls> / max(||patch_p||,1e-12); cls norm is a common
  // positive scale -> rank-invariant for top-k, omitted.
  for (int p = wv; p < PP; p += 8) {
    const float4* row = (const float4*)(xb + (size_t)(1 + p) * DD);
    const float4* cl  = (const float4*)clsL;
    float dot = 0.f, nrm = 0.f;
    for (int q = lane; q < DD / 4; q += 32) {
      float4 v = row[q]; float4 c = cl[q];
      dot += v.x * c.x + v.y * c.y + v.z * c.z + v.w * c.w;
      nrm += v.x * v.x + v.y * v.y + v.z * v.z + v.w * v.w;
    }
    for (int off = 16; off > 0; off >>= 1) {
      dot += __shfl_down(dot, off);
      nrm += __shfl_down(nrm, off);
    }
    if (lane == 0) simv[p] = dot / fmaxf(sqrtf(nrm), 1e-12f);
  }
  __syncthreads();

  // 10 rounds of parallel argmin (tie -> lower index, matches jax top_k)
  for (int k = 0; k < KOCC; ++k) {
    float bv = FLT_MAX; int bi = 0x7fffffff;
    for (int p = t; p < PP; p += 256) {
      float v = simv[p];
      if (v < bv) { bv = v; bi = p; }
    }
    redv[t] = bv; redi[t] = bi;
    __syncthreads();
    for (int s = 128; s > 0; s >>= 1) {
      if (t < s) {
        float v2 = redv[t + s]; int i2 = redi[t + s];
        if (v2 < redv[t] || (v2 == redv[t] && i2 < redi[t])) {
          redv[t] = v2; redi[t] = i2;
        }
      }
      __syncthreads();
    }
    if (t == 0) { maskv[redi[0]] = 0.0f; simv[redi[0]] = FLT_MAX; }
    __syncthreads();
  }
  for (int p = t; p < PP; p += 256) maskws[(size_t)b * PP + p] = maskv[p];
}

// Issue one tile's worth (16x768 f32) of async global->LDS b128 copies.
// Each wave contributes 12 instructions (ASYNCcnt += 12 per wave).
__device__ __forceinline__ void prefetch_tile(const float* __restrict__ xb,
                                              float* __restrict__ buf,
                                              int p0, int t) {
#pragma unroll
  for (int i = 0; i < 12; ++i) {
    int idx = t + i * 256;
    int row = idx / 192;
    int d   = (idx - row * 192) * 4;
    const float* g = xb + (size_t)(1 + p0 + row) * DD + d;
    // flat LDS addresses carry the LDS byte offset in addr[31:0] (ISA 10.2)
    unsigned lds = (unsigned)(uintptr_t)(buf + row * AS + d);
    asm volatile("global_load_async_to_lds_b128 %0, %1, off"
                 :: "v"(lds), "v"(g) : "memory");
  }
}

// ------------------------------------------------------------------
// Kernel 2: double-buffered async tile staging; WMMA scores GEMM
// (W B-frags register-resident), softmax over 17 groups, WMMA
// group-feature GEMM with C accumulators held in registers.
// ------------------------------------------------------------------
__global__ __launch_bounds__(256)
void group_kernel(const float* __restrict__ x, const float* __restrict__ gw,
                  const float* __restrict__ maskws, float* __restrict__ out,
                  float* __restrict__ gfpart, int tilesPerSplit, int direct) {
  extern __shared__ float sm[];
  float* tiles   = sm;                  // 2 * 16*AS = 25216 floats
  float* scoresP = sm + 2 * TILEF;      // 8 waves * 16p * 16g = 2048
  float* Wc16    = scoresP + 2048;      // 768 (group-16 column of W, pre-scaled)
  float* s16red  = Wc16 + DD;           // 256
  float* attnL   = s16red + 256;        // 17 rows (g) * stride 17 = 289

  const int split = blockIdx.x, b = blockIdx.y;
  const int t = threadIdx.x, lane = t & 31, wv = t >> 5;
  const int lm = lane & 15, hi = lane >> 4;
  const float* xb = x + (size_t)b * NN * DD;

  // W column 16 (scalar path) -> LDS, pre-scaled by 1/TEMP
  for (int i = t; i < DD; i += 256) Wc16[i] = gw[i * GG + 16] * INVTEMP;

  // GEMM1 B-fragments are loop-invariant per (wave,lane): preload the 48
  // W values this lane ever needs into registers (frees 52KB of LDS).
  float wreg[48];
  {
    const int base = wv * 96 + 2 * hi;
#pragma unroll
    for (int kk = 0; kk < 24; ++kk) {
      wreg[2 * kk]     = gw[(base + 4 * kk)     * GG + lm] * INVTEMP;
      wreg[2 * kk + 1] = gw[(base + 4 * kk + 1) * GG + lm] * INVTEMP;
    }
  }

  v8f cGF[6];
  for (int j = 0; j < 6; ++j) cGF[j] = {};
  float acc16[3] = {0.f, 0.f, 0.f};

  // kick off tile 0 into buffer 0
  prefetch_tile(xb, tiles, (split * tilesPerSplit) * 16, t);

  for (int ti = 0; ti < tilesPerSplit; ++ti) {
    const int p0 = (split * tilesPerSplit + ti) * 16;
    float* cur = tiles + (ti & 1) * TILEF;

    // prefetch next tile into the other buffer (its previous readers all
    // passed the end-of-iteration barrier below)
    if (ti + 1 < tilesPerSplit) {
      prefetch_tile(xb, tiles + ((ti + 1) & 1) * TILEF, p0 + 16, t);
      // 12 new outstanding; in-order completion => waiting to <=12 drains tile ti
      asm volatile("s_wait_asynccnt 0xc" ::: "memory");
    } else {
      asm volatile("s_wait_asynccnt 0x0" ::: "memory");
    }
    __syncthreads();   // all waves' async writes for tile ti are in LDS

    // GEMM1: scores[p, g=0..15] = tile(16x768) x W(768x16); K split over 8
    // waves; B comes from registers, A from LDS (conflict-free, AS%64==20).
    {
      v8f c = {};
      const int kbase = wv * 96;
#pragma unroll
      for (int kk = 0; kk < 24; ++kk) {
        int k = kbase + kk * 4;
        v2f a, bf;
        a[0]  = cur[lm * AS + k + 2 * hi];      // A 16x4 f32: v0 = K0|K2
        a[1]  = cur[lm * AS + k + 2 * hi + 1];  //             v1 = K1|K3
        bf[0] = wreg[2 * kk];
        bf[1] = wreg[2 * kk + 1];
        c = __builtin_amdgcn_wmma_f32_16x16x4_f32(
                false, a, false, bf, (short)0, c, false, false);
      }
      for (int r = 0; r < 8; ++r)
        scoresP[(wv * 16 + r + 8 * hi) * 16 + lm] = c[r];

      // 17th group column scalar (avoids a 94%-padded WMMA N-tile)
      int p = t >> 4, dp = t & 15;
      float s = 0.f;
      for (int d = dp; d < DD; d += 16) s += cur[p * AS + d] * Wc16[d];
      s16red[t] = s;
    }
    __syncthreads();

    // softmax over the 17 groups per patch; mask applied here (scores of
    // occluded patches are exactly 0 -> uniform 1/17 attn, matching ref),
    // and attnL is pre-multiplied by mask so GEMM2 can consume the RAW tile.
    if (t < 16) {
      const int p = t;
      const float m = maskws[(size_t)b * PP + p0 + p];   // 0 or 1
      float s[GG];
      for (int g = 0; g < 16; ++g) {
        float acc = 0.f;
        for (int w = 0; w < 8; ++w) acc += scoresP[w * 256 + p * 16 + g];
        s[g] = acc * m;
      }
      float s16 = 0.f;
      for (int j = 0; j < 16; ++j) s16 += s16red[p * 16 + j];
      s[16] = s16 * m;
      float mx = s[0];
      for (int g = 1; g < GG; ++g) mx = fmaxf(mx, s[g]);
      float sum = 0.f;
      for (int g = 0; g < GG; ++g) { s[g] = expf(s[g] - mx); sum += s[g]; }
      float inv = 1.0f / sum;
      float* ao = out + FEATS + (size_t)b * GG * PP + p0 + p;
      for (int g = 0; g < GG; ++g) {
        float a = s[g] * inv;
        attnL[g * 17 + p] = a * m;     // masked attn^T for GEMM2
        ao[(size_t)g * PP] = a;        // attn output (B,17,576)
      }
    }
    __syncthreads();

    // GEMM2: gf[g=0..15, d] += (attn*mask)^T(16x16) x raw tile(16x768)
    for (int kp = 0; kp < 16; kp += 4) {
      v2f a;
      a[0] = attnL[lm * 17 + kp + 2 * hi];
      a[1] = attnL[lm * 17 + kp + 2 * hi + 1];
#pragma unroll
      for (int j = 0; j < 6; ++j) {
        int d0 = (wv * 6 + j) * 16;
        v2f bf;
        bf[0] = cur[(kp + 2 * hi) * AS + d0 + lm];
        bf[1] = cur[(kp + 2 * hi + 1) * AS + d0 + lm];
        cGF[j] = __builtin_amdgcn_wmma_f32_16x16x4_f32(
                     false, a, false, bf, (short)0, cGF[j], false, false);
      }
    }
    // g = 16 row, scalar accumulate (each thread owns 3 d-slots)
    for (int i = 0; i < 3; ++i) {
      int d = t + 256 * i;
      float s = 0.f;
      for (int p = 0; p < 16; ++p) s += attnL[16 * 17 + p] * cur[p * AS + d];
      acc16[i] += s;
    }
    __syncthreads();   // everyone done with cur before it is re-prefetched
  }

  // write group features: direct into feats, or deterministic partials
  if (direct) {
    float* fb = out + (size_t)b * 18 * DD;
    for (int j = 0; j < 6; ++j) {
      int d = (wv * 6 + j) * 16 + lm;
      for (int r = 0; r < 8; ++r)
        fb[(size_t)(1 + r + 8 * hi) * DD + d] = cGF[j][r];
    }
    for (int i = 0; i < 3; ++i) fb[(size_t)17 * DD + t + 256 * i] = acc16[i];
  } else {
    float* pb = gfpart + (size_t)(split * BB + b) * GG * DD;
    for (int j = 0; j < 6; ++j) {
      int d = (wv * 6 + j) * 16 + lm;
      for (int r = 0; r < 8; ++r)
        pb[(size_t)(r + 8 * hi) * DD + d] = cGF[j][r];
    }
    for (int i = 0; i < 3; ++i) pb[(size_t)16 * DD + t + 256 * i] = acc16[i];
  }
}

// ------------------------------------------------------------------
// Kernel 3: deterministic reduction of per-split partial group features
// ------------------------------------------------------------------
__global__ __launch_bounds__(256)
void reduce_kernel(const float* __restrict__ gfpart, float* __restrict__ out,
                   int nsplit) {
  const int per = GG * DD;               // 13056
  int idx = blockIdx.x * 256 + threadIdx.x;
  if (idx >= BB * per) return;
  int b = idx / per;
  int rem = idx - b * per;
  float s = 0.f;
  for (int sp = 0; sp < nsplit; ++sp)
    s += gfpart[(size_t)(sp * BB + b) * per + rem];
  out[(size_t)b * 18 * DD + DD + rem] = s; // feats rows 1..17
}

extern "C" void kernel_launch(void* const* d_in, const int* in_sizes, int n_in,
                              void* d_out, int out_size, void* d_ws, size_t ws_size,
                              hipStream_t stream) {
  const float* x  = (const float*)d_in[0];   // (128,577,768) f32
  const float* gw = (const float*)d_in[1];   // (768,17) f32
  float* out = (float*)d_out;                // feats (128,18,768) ++ attn (128,17,576)
  float* maskws = (float*)d_ws;

  const size_t maskBytes = (size_t)BB * PP * sizeof(float);       // 294 KB
  const size_t perPart   = (size_t)BB * GG * DD * sizeof(float);  // 6.7 MB

  int nsplit = 1, direct = 1;
  float* gfpart = nullptr;
  if (ws_size >= maskBytes + 4 * perPart) {   // 4-way patch-tile split fills WGPs
    nsplit = 4; direct = 0;
    gfpart = (float*)((char*)d_ws + maskBytes);
  }

  prep_kernel<<<BB, 256, 0, stream>>>(x, out, maskws);

  const size_t smemB =
      (size_t)(2 * TILEF + 2048 + DD + 256 + 17 * 17 + 16) * sizeof(float);
  dim3 g2(nsplit, BB);
  group_kernel<<<g2, 256, smemB, stream>>>(x, gw, maskws, out, gfpart,
                                           36 / nsplit, direct);
  if (!direct) {
    int total = BB * GG * DD;
    reduce_kernel<<<(total + 255) / 256, 256, 0, stream>>>(gfpart, out, nsplit);
  }
}